// GraphormerMultiHeadAttention_66297115181077
// MI455X (gfx1250) — compile-verified
//
#include <hip/hip_runtime.h>
#include <hip/hip_bf16.h>
#include <cstdint>

// ---------------------------------------------------------------------------
// Graphormer MHA, fused per-d-slice attention for MI455X (gfx1250, wave32).
// N=1024, D=128, H=8.  All math f32 via V_WMMA_F32_16X16X4_F32.
// v_d staging uses GLOBAL_LOAD_ASYNC_TO_LDS_B128 (ASYNCcnt-tracked DMA).
//
// Workspace layout (floats): Q[1M] | K[1M] | V[1M] | Bias[1M] | Out2[1M]
//   = 20 MB total.
// ---------------------------------------------------------------------------

typedef __attribute__((ext_vector_type(2))) float v2f;
typedef __attribute__((ext_vector_type(8))) float v8f;

#define NN   1024
#define DD   128
#define HH   8

__device__ __forceinline__ v8f wmma4(v2f a, v2f b, v8f c) {
  // D = A(16x4) * B(4x16) + C(16x16), f32.
  return __builtin_amdgcn_wmma_f32_16x16x4_f32(
      /*neg_a=*/false, a, /*neg_b=*/false, b,
      /*c_mod=*/(short)0, c, /*reuse_a=*/false, /*reuse_b=*/false);
}

// Fragment loader shared by A (MxK row-major) and B (NxK row-major, i.e. B^T).
// For f32 16x16x4: lane&15 selects the M/N row, lane>>4 selects K pair {0,1}
// vs {2,3}; the two K values are adjacent in memory -> one float2.
__device__ __forceinline__ v2f load_frag_nk(const float* __restrict__ base,
                                            int row_stride, int r0, int k0,
                                            int lane) {
  const float* p = base + (size_t)(r0 + (lane & 15)) * row_stride
                        + k0 + ((lane >> 4) << 1);
  v2f r = {p[0], p[1]};
  return r;
}

// Async DMA: 16 bytes global -> LDS, tracked by ASYNCcnt (no VGPR data path).
__device__ __forceinline__ void async_g2l_b128(uint32_t lds_byte_addr,
                                               const float* gsrc) {
  asm volatile("global_load_async_to_lds_b128 %0, %1, off"
               :: "v"(lds_byte_addr), "v"((uint64_t)(uintptr_t)gsrc)
               : "memory");
}

__device__ __forceinline__ void wait_asynccnt0() {
#if defined(__has_builtin)
#if __has_builtin(__builtin_amdgcn_s_wait_asynccnt)
  __builtin_amdgcn_s_wait_asynccnt(0);
#else
  asm volatile("s_wait_asynccnt 0x0" ::: "memory");
#endif
#else
  asm volatile("s_wait_asynccnt 0x0" ::: "memory");
#endif
}

// ---------------------------------------------------------------------------
// Bias = spatial + edge   (1M floats, float4-vectorized)
// ---------------------------------------------------------------------------
__global__ void bias_kernel(const float* __restrict__ sp,
                            const float* __restrict__ ed,
                            float* __restrict__ B) {
  int i = blockIdx.x * blockDim.x + threadIdx.x;   // 1024 blocks * 256 thr
  const float4* s4 = (const float4*)sp;
  const float4* e4 = (const float4*)ed;
  float4 a = s4[i], b = e4[i];
  ((float4*)B)[i] = make_float4(a.x + b.x, a.y + b.y, a.z + b.z, a.w + b.w);
}

// ---------------------------------------------------------------------------
// Q/K/V = x @ W^T   (1024x1024, K=128).  grid (64, 16, 3), block 128.
// Each wave: one 16x16 tile; block covers 16 rows x 64 cols.
// ---------------------------------------------------------------------------
__global__ void qkv_kernel(const float* __restrict__ x,
                           const float* __restrict__ Wq,
                           const float* __restrict__ Wk,
                           const float* __restrict__ Wv,
                           float* __restrict__ Q,
                           float* __restrict__ K,
                           float* __restrict__ V) {
  const int lane = threadIdx.x & 31;
  const int wave = threadIdx.x >> 5;
  const int i0 = blockIdx.x * 16;
  const int j0 = blockIdx.y * 64 + wave * 16;

  const float* W;
  float* out;
  if (blockIdx.z == 0)      { W = Wq; out = Q; }
  else if (blockIdx.z == 1) { W = Wk; out = K; }
  else                      { W = Wv; out = V; }

  v8f c = {};
  #pragma unroll 4
  for (int k0 = 0; k0 < DD; k0 += 4) {
    v2f a = load_frag_nk(x, DD, i0, k0, lane);   // x row-major [N, D]
    v2f b = load_frag_nk(W, DD, j0, k0, lane);   // W row-major [DH, D] = NxK
    c = wmma4(a, b, c);
  }

  const int row_half = (lane >> 4) << 3;     // lanes 16-31 hold rows 8..15
  const int col = lane & 15;
  #pragma unroll
  for (int r = 0; r < 8; ++r)
    out[(size_t)(i0 + r + row_half) * NN + j0 + col] = c[r];
}

// ---------------------------------------------------------------------------
// Fused per-slice attention.  grid (64 n-tiles, 128 slices), block 128.
// Dynamic LDS: S[16][1024] + vbuf[1024*8] + red[128] + cpart[4*256]
//            = 25728 floats = 102912 bytes (WGP LDS is 320 KB).
// ---------------------------------------------------------------------------
__global__ void attn_kernel(const float* __restrict__ Q,
                            const float* __restrict__ Km,
                            const float* __restrict__ V,
                            const float* __restrict__ Bias,
                            float* __restrict__ O2) {
  extern __shared__ float smem[];
  float* S     = smem;                 // 16384 : logits/probs, 16 rows x 1024
  float* vbuf  = smem + 16384;         //  8192 : v_d staged (async DMA)
  float* red   = vbuf + 8192;          //   128 : softmax partials
  float* cpart = red + 128;            //  1024 : split-K partial C tiles

  const int tid  = threadIdx.x;
  const int lane = tid & 31;
  const int wave = tid >> 5;
  const int i0   = blockIdx.x * 16;            // n-row tile
  const int d    = blockIdx.y;                 // slice

  const float* qd = Q  + (size_t)d * (NN * HH);   // contiguous 1024x8 block
  const float* kd = Km + (size_t)d * (NN * HH);
  const float* vd = V  + (size_t)d * (NN * HH);
  const float scale = 0.088388347648318447f;      // 128^-0.5

  // Stage v_d (32 KB) into LDS with async global->LDS DMA (ASYNCcnt).
  // Generic->LDS address: low 32 bits of an LDS-aperture address are the
  // LDS byte offset (ISA 10.2 aperture table).
  {
    const uint32_t vbase = (uint32_t)(uintptr_t)vbuf;
    #pragma unroll
    for (int i = tid; i < (NN * HH) / 4; i += 128)
      async_g2l_b128(vbase + (uint32_t)i * 16u, vd + i * 4);
    wait_asynccnt0();   // our slots visible in LDS before the barrier below
  }

  // Pre-scaled A fragments of q_d (16 rows, K=8 -> two K=4 fragments).
  v2f a0 = load_frag_nk(qd, HH, i0, 0, lane);
  v2f a1 = load_frag_nk(qd, HH, i0, 4, lane);
  a0 *= scale;
  a1 *= scale;

  const int row_half = (lane >> 4) << 3;
  const int col = lane & 15;

  // Logits: wave w owns columns [w*256, w*256+256).  D = (s*q)k^T + Bias.
  for (int t = 0; t < 16; ++t) {
    const int j0 = wave * 256 + t * 16;
    // Prefetch next tile's bias rows into L0 while this tile computes.
    if (t < 15)
      __builtin_prefetch(&Bias[(size_t)(i0 + row_half) * NN + j0 + 16 + col],
                         0, 3);
    v8f c;
    #pragma unroll
    for (int r = 0; r < 8; ++r)
      c[r] = Bias[(size_t)(i0 + r + row_half) * NN + j0 + col];
    v2f b0 = load_frag_nk(kd, HH, j0, 0, lane);  // k_d is NxK row-major
    v2f b1 = load_frag_nk(kd, HH, j0, 4, lane);
    c = wmma4(a0, b0, c);
    c = wmma4(a1, b1, c);
    #pragma unroll
    for (int r = 0; r < 8; ++r)
      S[(r + row_half) * NN + j0 + col] = c[r];
  }
  __syncthreads();

  // Softmax over the 1024 columns of each of the 16 rows.
  // 8 threads per row, each owns a contiguous 128-column chunk.
  {
    const int r = tid >> 3;
    const int sub = tid & 7;
    float* row = S + r * NN + sub * 128;

    float m = -3.0e38f;
    for (int c2 = 0; c2 < 128; ++c2) m = fmaxf(m, row[c2]);
    red[r * 8 + sub] = m;
    __syncthreads();
    float rowmax = red[r * 8];
    #pragma unroll
    for (int k = 1; k < 8; ++k) rowmax = fmaxf(rowmax, red[r * 8 + k]);
    __syncthreads();

    float s = 0.f;
    for (int c2 = 0; c2 < 128; ++c2) {
      float e = __expf(row[c2] - rowmax);
      row[c2] = e;
      s += e;
    }
    red[r * 8 + sub] = s;
    __syncthreads();
    float tot = 0.f;
    #pragma unroll
    for (int k = 0; k < 8; ++k) tot += red[r * 8 + k];
    const float inv = 1.0f / tot;
    for (int c2 = 0; c2 < 128; ++c2) row[c2] *= inv;
  }
  __syncthreads();

  // O = S @ v_d  (16x1024 @ 1024x8, padded to N=16).  Split-K across waves.
  v8f o = {};
  for (int k0 = wave * 256; k0 < wave * 256 + 256; k0 += 4) {
    const int kk = k0 + ((lane >> 4) << 1);
    v2f a = {S[(lane & 15) * NN + kk], S[(lane & 15) * NN + kk + 1]};
    float bx = (col < HH) ? vbuf[kk * HH + col] : 0.f;
    float by = (col < HH) ? vbuf[(kk + 1) * HH + col] : 0.f;
    v2f b = {bx, by};
    o = wmma4(a, b, o);
  }
  #pragma unroll
  for (int r = 0; r < 8; ++r)
    cpart[wave * 256 + (r + row_half) * 16 + col] = o[r];
  __syncthreads();

  // Reduce 4 partials; write 16x8 block into Out2 (flat reinterpretation:
  // out[d, n, h] -> Out2.flat[d*8192 + n*8 + h]).
  {
    const int r = tid >> 3;
    const int h = tid & 7;
    float v = cpart[r * 16 + h] + cpart[256 + r * 16 + h] +
              cpart[512 + r * 16 + h] + cpart[768 + r * 16 + h];
    O2[(size_t)d * (NN * HH) + (i0 + r) * HH + h] = v;
  }
}

// ---------------------------------------------------------------------------
// Y = Out2 @ Wo^T   (1024x1024 @ [128x1024]^T -> 1024x128).
// grid (64, 8), block 128: all 4 waves on one 16x16 tile, split-K, LDS reduce.
// ---------------------------------------------------------------------------
__global__ void oproj_kernel(const float* __restrict__ O2,
                             const float* __restrict__ Wo,
                             float* __restrict__ Y) {
  __shared__ float cpart[4 * 256];
  const int tid = threadIdx.x;
  const int lane = tid & 31;
  const int wave = tid >> 5;
  const int i0 = blockIdx.x * 16;
  const int j0 = blockIdx.y * 16;

  v8f c = {};
  #pragma unroll 4
  for (int k0 = wave * 256; k0 < wave * 256 + 256; k0 += 4) {
    v2f a = load_frag_nk(O2, NN, i0, k0, lane);  // [1024,1024] row-major
    v2f b = load_frag_nk(Wo, NN, j0, k0, lane);  // Wo [128,1024] = NxK
    c = wmma4(a, b, c);
  }

  const int row_half = (lane >> 4) << 3;
  const int col = lane & 15;
  #pragma unroll
  for (int r = 0; r < 8; ++r)
    cpart[wave * 256 + (r + row_half) * 16 + col] = c[r];
  __syncthreads();

  for (int idx = tid; idx < 256; idx += 128) {
    const int r = idx >> 4, cc = idx & 15;
    float v = cpart[r * 16 + cc] + cpart[256 + r * 16 + cc] +
              cpart[512 + r * 16 + cc] + cpart[768 + r * 16 + cc];
    Y[(size_t)(i0 + r) * DD + j0 + cc] = v;
  }
}

// ---------------------------------------------------------------------------
extern "C" void kernel_launch(void* const* d_in, const int* in_sizes, int n_in,
                              void* d_out, int out_size, void* d_ws,
                              size_t ws_size, hipStream_t stream) {
  const float* x  = (const float*)d_in[0];
  const float* sp = (const float*)d_in[1];
  const float* ed = (const float*)d_in[2];
  const float* Wq = (const float*)d_in[3];
  const float* Wk = (const float*)d_in[4];
  const float* Wv = (const float*)d_in[5];
  const float* Wo = (const float*)d_in[6];
  float* out = (float*)d_out;

  float* ws = (float*)d_ws;
  const size_t M1 = (size_t)NN * NN;       // 1M floats
  float* Q    = ws;
  float* K    = ws + 1 * M1;
  float* V    = ws + 2 * M1;
  float* Bias = ws + 3 * M1;
  float* O2   = ws + 4 * M1;               // needs 20 MB of workspace

  // 1) Bias = spatial + edge
  bias_kernel<<<dim3(M1 / (256 * 4)), 256, 0, stream>>>(sp, ed, Bias);

  // 2) Q, K, V projections
  qkv_kernel<<<dim3(NN / 16, NN / 64, 3), 128, 0, stream>>>(x, Wq, Wk, Wv,
                                                            Q, K, V);

  // 3) Fused per-slice attention (softmax in LDS)
  const size_t smem_bytes = (16384 + 8192 + 128 + 1024) * sizeof(float);
  attn_kernel<<<dim3(NN / 16, DD), 128, smem_bytes, stream>>>(Q, K, V, Bias,
                                                              O2);

  // 4) Output projection
  oproj_kernel<<<dim3(NN / 16, DD / 16), 128, 0, stream>>>(O2, Wo, out);
}